// ALNNLayer_4475355922752
// MI455X (gfx1250) — compile-verified
//
#include <hip/hip_runtime.h>

// ALNN layer for MI455X (gfx1250, wave32).
// Memory-bound fp32 problem (~14 MB working set, fits in 192 MB L2).
// L-reduction (K=128) is done on the matrix pipe: 32 chained
// V_WMMA_F32_16X16X4_F32 ops with an all-ones B matrix act as a free
// row-sum accumulator; 4 independent accumulator chains (4 batch rows)
// amortize weight loads and hide WMMA latency.

typedef float v2f __attribute__((ext_vector_type(2)));
typedef float v8f __attribute__((ext_vector_type(8)));

#define B_SZ 32
#define R_SZ 48
#define L_SZ 128
#define D_SZ 64
#define BQ   4   // batch rows per wave (independent WMMA accumulator chains)

__global__ __launch_bounds__(32)
void alnn_wmma_kernel(const float* __restrict__ X,
                      const float* __restrict__ T,
                      const float* __restrict__ Mk,
                      const float* __restrict__ PD,
                      const float* __restrict__ alpha,
                      const float* __restrict__ w_v,
                      const float* __restrict__ w_t,
                      const float* __restrict__ b_t,
                      const float* __restrict__ b_v,
                      const float* __restrict__ ref_time,
                      float* __restrict__ out)
{
    // wave -> (b-quad, r, d-block)
    const int wid  = blockIdx.x;
    const int dblk = wid & 3;                 // D_SZ/16 = 4 blocks
    const int r    = (wid >> 2) % R_SZ;
    const int b0   = (wid / (4 * R_SZ)) * BQ;

    const int lane = threadIdx.x;
    const int mrow = lane & 15;               // WMMA A-matrix row (M)
    const int khi  = lane >> 4;               // 0: K=0,1  1: K=2,3
    const int d    = dblk * 16 + mrow;

    const float a_r = fmaxf(alpha[r], 0.0f);  // relu(alpha)
    const float rt  = ref_time[r];

    v8f acc0 = {}, acc1 = {}, acc2 = {}, acc3 = {};
    v2f ones; ones[0] = 1.0f; ones[1] = 1.0f; // B matrix = all ones -> row sum

    const float4* __restrict__ wt4 = (const float4*)w_t;

    for (int lc = 0; lc < L_SZ / 4; ++lc) {
        v2f av0, av1, av2, av3;
#pragma unroll
        for (int k = 0; k < 2; ++k) {
            const int l  = lc * 4 + khi * 2 + k;       // this lane's l value
            const int wo = (r * L_SZ + l) * D_SZ + d;  // weight offset
            const float4 wt  = wt4[wo];                // w_t[r,l,d,0..3], 16B load
            const float  bt4 = 4.0f * b_t[wo];         // b_t broadcast over e-axis
            const float  wv  = w_v[wo];
#pragma unroll
            for (int bb = 0; bb < BQ; ++bb) {
                const int xo = ((b0 + bb) * L_SZ + l) * D_SZ + d;
                const float x    = X[xo];
                const float dist = fabsf(T[xo] - rt);
                const float kern = __expf(-a_r * dist);
                const float inten = fmaxf(x * kern, 0.0f);
                float lat = wt.x * x + wt.y * inten + wt.z * Mk[xo]
                          + wt.w * PD[xo] + bt4;
                lat = fmaxf(lat, 0.0f);
                const float a = lat * wv;              // fold w_v into A
                if      (bb == 0) av0[k] = a;
                else if (bb == 1) av1[k] = a;
                else if (bb == 2) av2[k] = a;
                else              av3[k] = a;
            }
        }
        // D = A(16x4) * ones(4x16) + C  -> every column = row-sum over 4 l's
        acc0 = __builtin_amdgcn_wmma_f32_16x16x4_f32(false, av0, false, ones,
                                                     (short)0, acc0, false, false);
        acc1 = __builtin_amdgcn_wmma_f32_16x16x4_f32(false, av1, false, ones,
                                                     (short)0, acc1, false, false);
        acc2 = __builtin_amdgcn_wmma_f32_16x16x4_f32(false, av2, false, ones,
                                                     (short)0, acc2, false, false);
        acc3 = __builtin_amdgcn_wmma_f32_16x16x4_f32(false, av3, false, ones,
                                                     (short)0, acc3, false, false);
    }

    // C/D layout: VGPR v, lanes 0-15 -> M=v (col N=lane); lanes 16-31 -> M=v+8.
    // Column N==0 lives in lane 0 (M=0..7) and lane 16 (M=8..15).
    if (mrow == 0) {
#pragma unroll
        for (int v = 0; v < 8; ++v) {
            const int   dd   = dblk * 16 + khi * 8 + v;
            const float bias = (float)L_SZ * b_v[r * D_SZ + dd];  // b_v bcast over L
            const float r0 = fmaxf(acc0[v] + bias, 0.0f);
            const float r1 = fmaxf(acc1[v] + bias, 0.0f);
            const float r2 = fmaxf(acc2[v] + bias, 0.0f);
            const float r3 = fmaxf(acc3[v] + bias, 0.0f);
            out[((b0 + 0) * R_SZ + r) * D_SZ + dd] = r0;
            out[((b0 + 1) * R_SZ + r) * D_SZ + dd] = r1;
            out[((b0 + 2) * R_SZ + r) * D_SZ + dd] = r2;
            out[((b0 + 3) * R_SZ + r) * D_SZ + dd] = r3;
        }
    }
}

extern "C" void kernel_launch(void* const* d_in, const int* in_sizes, int n_in,
                              void* d_out, int out_size, void* d_ws, size_t ws_size,
                              hipStream_t stream) {
    (void)in_sizes; (void)n_in; (void)d_ws; (void)ws_size; (void)out_size;
    const float* X        = (const float*)d_in[0];
    const float* T        = (const float*)d_in[1];
    const float* Mk       = (const float*)d_in[2];
    const float* PD       = (const float*)d_in[3];
    const float* alpha    = (const float*)d_in[4];
    const float* w_v      = (const float*)d_in[5];
    const float* w_t      = (const float*)d_in[6];
    const float* b_t      = (const float*)d_in[7];
    const float* b_v      = (const float*)d_in[8];
    const float* ref_time = (const float*)d_in[9];
    float* out = (float*)d_out;

    const int n_waves = (B_SZ / BQ) * R_SZ * (D_SZ / 16);  // 8*48*4 = 1536
    alnn_wmma_kernel<<<n_waves, 32, 0, stream>>>(X, T, Mk, PD, alpha, w_v,
                                                 w_t, b_t, b_v, ref_time, out);
}